// EfficientDet_58239756534308
// MI455X (gfx1250) — compile-verified
//
#include <hip/hip_runtime.h>
#include <hip/hip_bf16.h>
#include <stdint.h>

// ---------------- problem constants (match reference) ----------------
#define BATCH       8
#define NANCH       49104
#define NCLS        90
#define POST_NMS    100
#define IOU_THR     0.5f
#define LOGIT_THR  (-1.3862943611f)   // log(0.2/0.8): score>0.2 in logit space
#define IMG_INV    (1.0f/512.0f)

// ---------------- kernel-tuning constants ----------------
#define THREADS     256
#define NWAVES      (THREADS/32)
#define WRANGE      (NANCH/NWAVES)    // 6138, exact
#define CAND        2048              // candidate cap per (b,c); >> 100 NMS picks
#define NBINS       128
#define HIST_LO     LOGIT_THR
#define HIST_HI     12.0f
#define NEG_INF    (-1.0e30f)

typedef unsigned int u32x4 __attribute__((ext_vector_type(4)));
typedef int          i32x8 __attribute__((ext_vector_type(8)));
typedef int          i32x4 __attribute__((ext_vector_type(4)));

// =====================================================================
// Kernel 1: decode anchors + deltas -> normalized clipped boxes [B,N,4]
// Anchors derived analytically from the flat index: zero anchor-table
// memory traffic.
// =====================================================================
__device__ __forceinline__ void anchor_of(int i, float& cy, float& cx,
                                          float& hh, float& hw) {
    int off, feat; float stride;
    if      (i < 36864) { off = 0;     feat = 64; stride = 8.0f;   }
    else if (i < 46080) { off = 36864; feat = 32; stride = 16.0f;  }
    else if (i < 48384) { off = 46080; feat = 16; stride = 32.0f;  }
    else if (i < 48960) { off = 48384; feat = 8;  stride = 64.0f;  }
    else                { off = 48960; feat = 4;  stride = 128.0f; }
    int r    = i - off;
    int cell = r / 9;
    int k    = r - cell * 9;
    int y    = cell / feat;
    int x    = cell - y * feat;
    cy = ((float)y + 0.5f) * stride;
    cx = ((float)x + 0.5f) * stride;
    const float p2[3] = {1.0f, 1.2599210499f, 1.5874010520f}; // 2^(i/3)
    const float sq[3] = {1.0f, 0.7071067812f, 1.4142135624f}; // sqrt(ratio)
    int si = k / 3;
    int rj = k - si * 3;
    float base = p2[si] * stride * 4.0f;
    hh = base / sq[rj] * 0.5f;
    hw = base * sq[rj] * 0.5f;
}

__global__ void __launch_bounds__(THREADS)
effdet_decode_kernel(const float* __restrict__ deltas,   // [B,N,4]
                     float* __restrict__ boxes)          // [B,N,4] normalized
{
    int t = blockIdx.x * THREADS + threadIdx.x;
    if (t >= BATCH * NANCH) return;
    int i = t % NANCH;

    float cy, cx, hh, hw;
    anchor_of(i, cy, cx, hh, hw);
    float ah = 2.0f * hh, aw = 2.0f * hw;

    float4 d = reinterpret_cast<const float4*>(deltas)[t];
    float ncy = d.x * ah + cy;
    float ncx = d.y * aw + cx;
    float h   = __expf(d.z) * ah;
    float w   = __expf(d.w) * aw;

    float y1 = (ncy - 0.5f * h) * IMG_INV;
    float x1 = (ncx - 0.5f * w) * IMG_INV;
    float y2 = (ncy + 0.5f * h) * IMG_INV;
    float x2 = (ncx + 0.5f * w) * IMG_INV;
    y1 = fminf(fmaxf(y1, 0.0f), 1.0f);
    x1 = fminf(fmaxf(x1, 0.0f), 1.0f);
    y2 = fminf(fmaxf(y2, 0.0f), 1.0f);
    x2 = fminf(fmaxf(x2, 0.0f), 1.0f);
    reinterpret_cast<float4*>(boxes)[t] = make_float4(y1, x1, y2, x2);
}

// =====================================================================
// Kernel 2: per-(batch,class) candidate selection + greedy NMS in LDS.
// One workgroup (8 wave32s) per (b,c); 720 workgroups.
//  - histogram-select ~top-CAND logits; boundary bin filled deterministically
//    in global index order via per-wave contiguous ranges + wave32 ballot
//  - candidate boxes gathered with CDNA5 async global->LDS b128 loads
//  - 100 iterations of argmax-reduce + IoU suppression, all in LDS
// =====================================================================
__device__ __forceinline__ int bin_of(float v, float scale) {
    int bn = (int)((v - HIST_LO) * scale);
    return bn < 0 ? 0 : (bn >= NBINS ? NBINS - 1 : bn);
}

__global__ void __launch_bounds__(THREADS)
effdet_nms_kernel(const float* __restrict__ logits,   // [B,N,C]
                  const float* __restrict__ boxes,    // [B,N,4] normalized
                  float* __restrict__ sel_box,        // [B,C,100,4]
                  float* __restrict__ sel_s)          // [B,C,100]
{
    __shared__ float cbox[CAND][4];
    __shared__ float csc[CAND];
    __shared__ int   cidx[CAND];
    __shared__ int   hist[NBINS];
    __shared__ float rsc[THREADS];
    __shared__ int   rai[THREADS];
    __shared__ int   rslot[THREADS];
    __shared__ int   wcnt[NWAVES];
    __shared__ int   wbase[NWAVES];
    __shared__ int   ctr[4];   // 0: strict-above count, 1: cutoff bin, 2: total

    const int wg   = blockIdx.x;
    const int b    = wg / NCLS;
    const int c    = wg - b * NCLS;
    const int tid  = threadIdx.x;
    const int wave = tid >> 5;
    const int lane = tid & 31;

    const float* col = logits + (size_t)b * NANCH * NCLS + c;
    const float bin_scale = (float)NBINS / (HIST_HI - HIST_LO);

    // ---- phase A: histogram of above-threshold logits ----
    for (int j = tid; j < NBINS; j += THREADS) hist[j] = 0;
    if (tid < 4) ctr[tid] = 0;
    if (tid < NWAVES) wcnt[tid] = 0;
    __syncthreads();

    for (int i = tid; i < NANCH; i += THREADS) {
        float v = col[(size_t)i * NCLS];
        if (i + 8 * THREADS < NANCH)
            __builtin_prefetch(&col[(size_t)(i + 8 * THREADS) * NCLS], 0, 0);
        if (v > LOGIT_THR) atomicAdd(&hist[bin_of(v, bin_scale)], 1);
    }
    __syncthreads();

    // ---- phase B: find cutoff bin (thread 0) ----
    if (tid == 0) {
        int cum = 0, cb = -1;
        for (int bn = NBINS - 1; bn >= 0; --bn) {
            int h = hist[bn];
            if (cum + h >= CAND) { cb = bn; break; }
            cum += h;
        }
        ctr[1] = cb;
    }
    __syncthreads();
    const int cutbin = ctr[1];

    // ---- phase C1: append everything strictly above the cutoff bin,
    //      and count boundary-bin elements per wave (contiguous ranges) ----
    const int range_lo = wave * WRANGE;
    const int range_hi = (wave == NWAVES - 1) ? NANCH : range_lo + WRANGE;
    {
        int bcnt = 0;
        // strict-above append over the whole grid-stride range
        for (int i = tid; i < NANCH; i += THREADS) {
            float v = col[(size_t)i * NCLS];
            if (v > LOGIT_THR) {
                int bn = bin_of(v, bin_scale);
                if (cutbin < 0 || bn > cutbin) {
                    int p = atomicAdd(&ctr[0], 1);
                    if (p < CAND) { csc[p] = v; cidx[p] = i; }
                }
            }
        }
        // boundary count over this wave's contiguous range
        if (cutbin >= 0) {
            for (int i = range_lo + lane; i < range_hi; i += 32) {
                float v = col[(size_t)i * NCLS];
                if (v > LOGIT_THR && bin_of(v, bin_scale) == cutbin) ++bcnt;
            }
            atomicAdd(&wcnt[wave], bcnt);
        }
    }
    __syncthreads();

    // ---- phase C2: prefix over waves, then ordered ballot append ----
    if (cutbin >= 0) {
        if (tid == 0) {
            int basep = ctr[0];
            for (int w = 0; w < NWAVES; ++w) { wbase[w] = basep; basep += wcnt[w]; }
            ctr[2] = basep;
        }
        __syncthreads();
        int run = wbase[wave];
        for (int i0 = range_lo; i0 < range_hi && run < CAND; i0 += 32) {
            int  i = i0 + lane;
            bool p = false;
            float v = NEG_INF;
            if (i < range_hi) {
                v = col[(size_t)i * NCLS];
                p = (v > LOGIT_THR) && (bin_of(v, bin_scale) == cutbin);
            }
            unsigned m   = (unsigned)__ballot(p);     // wave32: low 32 bits
            int      pre = __popc(m & ((1u << lane) - 1u));
            int      slot = run + pre;
            if (p && slot < CAND) { csc[slot] = v; cidx[slot] = i; }
            run += __popc(m);
        }
    } else {
        if (tid == 0) ctr[2] = ctr[0];
    }
    __syncthreads();
    const int ncand = ctr[2] < CAND ? ctr[2] : CAND;

    // ---- pad unused slots ----
    for (int j = tid; j < CAND; j += THREADS) {
        if (j >= ncand) { csc[j] = NEG_INF; cidx[j] = 0x7fffffff; }
    }
    __syncthreads();

    // ---- phase D: async global->LDS gather of candidate boxes (CDNA5) ----
    {
        const float* gb = boxes + (size_t)b * NANCH * 4;
        for (int j = tid; j < ncand; j += THREADS) {
            unsigned lds_off = (unsigned)(uintptr_t)(&cbox[j][0]);
            unsigned long long ga =
                (unsigned long long)(uintptr_t)(gb + (size_t)cidx[j] * 4);
            asm volatile("global_load_async_to_lds_b128 %0, %1, off"
                         :: "v"(lds_off), "v"(ga) : "memory");
        }
        asm volatile("s_wait_asynccnt 0" ::: "memory");
    }
    __syncthreads();

    // ---- phase E: greedy NMS, 100 selections ----
    float* sb = sel_box + (size_t)wg * POST_NMS * 4;
    float* ss = sel_s   + (size_t)wg * POST_NMS;

    for (int t = 0; t < POST_NMS; ++t) {
        // per-thread argmax (tie-break: lower anchor index)
        float bsc  = csc[tid];
        int   bai  = cidx[tid];
        int   bslt = tid;
        for (int j = tid + THREADS; j < CAND; j += THREADS) {
            float v = csc[j];
            int   a = cidx[j];
            if (v > bsc || (v == bsc && a < bai)) { bsc = v; bai = a; bslt = j; }
        }
        rsc[tid] = bsc; rai[tid] = bai; rslot[tid] = bslt;
        __syncthreads();
        for (int s = THREADS / 2; s > 0; s >>= 1) {
            if (tid < s) {
                float v = rsc[tid + s];
                int   a = rai[tid + s];
                if (v > rsc[tid] || (v == rsc[tid] && a < rai[tid])) {
                    rsc[tid] = v; rai[tid] = a; rslot[tid] = rslot[tid + s];
                }
            }
            __syncthreads();
        }
        int   best  = rslot[0];
        float bestv = rsc[0];

        if (!(bestv > LOGIT_THR)) {
            for (int j = t + tid; j < POST_NMS; j += THREADS) {
                ss[j] = 0.0f;
                sb[4 * j + 0] = 0.0f; sb[4 * j + 1] = 0.0f;
                sb[4 * j + 2] = 0.0f; sb[4 * j + 3] = 0.0f;
            }
            break;
        }

        float by1 = cbox[best][0], bx1 = cbox[best][1];
        float by2 = cbox[best][2], bx2 = cbox[best][3];
        if (tid == 0) {
            ss[t] = 1.0f / (1.0f + __expf(-bestv));   // sigmoid only at output
            sb[4 * t + 0] = by1; sb[4 * t + 1] = bx1;
            sb[4 * t + 2] = by2; sb[4 * t + 3] = bx2;
        }

        // suppress IoU > 0.5 (self-suppression via IoU == 1)
        float barea = (by2 - by1) * (bx2 - bx1);
        for (int j = tid; j < CAND; j += THREADS) {
            float v = csc[j];
            if (v <= -1.0e29f) continue;
            float yy1 = fmaxf(by1, cbox[j][0]);
            float xx1 = fmaxf(bx1, cbox[j][1]);
            float yy2 = fminf(by2, cbox[j][2]);
            float xx2 = fminf(bx2, cbox[j][3]);
            float inter = fmaxf(yy2 - yy1, 0.0f) * fmaxf(xx2 - xx1, 0.0f);
            float a2 = (cbox[j][2] - cbox[j][0]) * (cbox[j][3] - cbox[j][1]);
            float iou = inter / (barea + a2 - inter + 1e-8f);
            if (iou > IOU_THR) csc[j] = NEG_INF;
        }
        __syncthreads();
    }
}

// =====================================================================
// Kernel 3: per-batch final top-100 over the 90*100 NMS survivors.
// The contiguous 9000-float score block is staged into LDS with a TDM
// bulk transfer (tensor_load_to_lds + s_wait_tensorcnt).
// =====================================================================
__global__ void __launch_bounds__(THREADS)
effdet_topk_kernel(const float* __restrict__ sel_box,  // [B,C,100,4]
                   const float* __restrict__ sel_s,    // [B,C,100]
                   float* __restrict__ out)            // concat outputs
{
    const int FLAT = NCLS * POST_NMS;   // 9000
    __shared__ float s[NCLS * POST_NMS];
    __shared__ float rsc[THREADS];
    __shared__ int   rid[THREADS];

    const int b   = blockIdx.x;
    const int tid = threadIdx.x;

    const float* ss_in = sel_s + (size_t)b * FLAT;

    // ---- TDM bulk load: 9000 contiguous f32 -> LDS ----
    if (tid == 0) {
        unsigned long long ga = (unsigned long long)(uintptr_t)ss_in;
        unsigned lds_off = (unsigned)(uintptr_t)(&s[0]);
        u32x4 g0;
        g0.x = 1u;                                     // count=1, user descriptor
        g0.y = lds_off;                                // lds_addr
        g0.z = (unsigned)(ga & 0xffffffffu);           // global_addr[31:0]
        g0.w = (unsigned)((ga >> 32) & 0x01ffffffu)    // global_addr[56:32]
             | (2u << 30);                             // type=2 (image/D#)
        i32x8 g1;
        g1[0] = (2 << 16);                             // data_size=4B, no flags
        g1[1] = (int)((unsigned)(FLAT & 0xffff) << 16);// tensor_dim0[15:0]
        g1[2] = (FLAT >> 16) | (1 << 16);              // dim0 hi | tensor_dim1=1
        g1[3] = (int)((unsigned)(FLAT & 0xffff) << 16);// tile_dim0 = 9000
        g1[4] = 1;                                     // tile_dim1=1, tile_dim2=0
        g1[5] = FLAT;                                  // tensor_dim0_stride lo
        g1[6] = 0;                                     // stride hi | dim1_stride lo
        g1[7] = 0;
        i32x4 g2 = {0, 0, 0, 0};                       // group 2 (unused, 2-D)
        i32x4 g3 = {0, 0, 0, 0};                       // group 3 (unused, 2-D)
        i32x8 g4 = {0, 0, 0, 0, 0, 0, 0, 0};           // trailing group (unused)
        __builtin_amdgcn_tensor_load_to_lds(g0, g1, g2, g3, g4, 0);
        __builtin_amdgcn_s_wait_tensorcnt(0);
    }
    __syncthreads();

    float* ob = out;                                    // [B,100,4]
    float* os = out + (size_t)BATCH * POST_NMS * 4;     // [B,100]
    float* oc = os  + (size_t)BATCH * POST_NMS;         // [B,100]
    float* ov = oc  + (size_t)BATCH * POST_NMS;         // [B]

    int nvalid = 0;
    for (int t = 0; t < POST_NMS; ++t) {
        float bs = NEG_INF;
        int   bi = 0x7fffffff;
        for (int j = tid; j < FLAT; j += THREADS) {
            float v = s[j];
            if (v > bs || (v == bs && j < bi)) { bs = v; bi = j; }
        }
        rsc[tid] = bs; rid[tid] = bi;
        __syncthreads();
        for (int st = THREADS / 2; st > 0; st >>= 1) {
            if (tid < st) {
                if (rsc[tid + st] > rsc[tid] ||
                    (rsc[tid + st] == rsc[tid] && rid[tid + st] < rid[tid])) {
                    rsc[tid] = rsc[tid + st];
                    rid[tid] = rid[tid + st];
                }
            }
            __syncthreads();
        }
        if (tid == 0) {
            int   bidx = rid[0];
            float bval = rsc[0];
            os[(size_t)b * POST_NMS + t] = bval;
            oc[(size_t)b * POST_NMS + t] = (float)(bidx / POST_NMS);
            const float* bb = sel_box + ((size_t)b * FLAT + bidx) * 4;
            float* obt = ob + ((size_t)b * POST_NMS + t) * 4;
            obt[0] = bb[0]; obt[1] = bb[1]; obt[2] = bb[2]; obt[3] = bb[3];
            s[bidx] = NEG_INF;
            if (bval > 0.0f) ++nvalid;
        }
        __syncthreads();
    }
    if (tid == 0) ov[b] = (float)nvalid;
}

// =====================================================================
// launch
// =====================================================================
extern "C" void kernel_launch(void* const* d_in, const int* in_sizes, int n_in,
                              void* d_out, int out_size, void* d_ws, size_t ws_size,
                              hipStream_t stream) {
    const float* box_deltas = (const float*)d_in[0];  // [B,N,4]
    const float* cls_logits = (const float*)d_in[1];  // [B,N,C]
    float* out = (float*)d_out;

    // workspace layout (floats)
    float* ws       = (float*)d_ws;
    float* boxes    = ws;                                     // B*N*4
    float* sel_box  = boxes   + (size_t)BATCH * NANCH * 4;    // B*C*100*4
    float* sel_s    = sel_box + (size_t)BATCH * NCLS * POST_NMS * 4; // B*C*100

    (void)in_sizes; (void)n_in; (void)out_size; (void)ws_size;

    int decode_blocks = (BATCH * NANCH + THREADS - 1) / THREADS;
    effdet_decode_kernel<<<decode_blocks, THREADS, 0, stream>>>(box_deltas, boxes);

    effdet_nms_kernel<<<BATCH * NCLS, THREADS, 0, stream>>>(
        cls_logits, boxes, sel_box, sel_s);

    effdet_topk_kernel<<<BATCH, THREADS, 0, stream>>>(sel_box, sel_s, out);
}